// ALTER2Layer_75840532512812
// MI455X (gfx1250) — compile-verified
//
#include <hip/hip_runtime.h>
#include <hip/hip_bf16.h>

#define DIMN   1024
#define C0N    64
#define C1N    32
#define BATCHN 128
#define DTILE  128
#define B2STR  144   // padded row stride (floats) -> bank-conflict-free B-frag reads
#define JACELTS 134217728ull        // 128*1024*1024
#define JACOFF  135168u             // 128*1024 + 128*32

typedef __attribute__((ext_vector_type(2))) float v2f;
typedef __attribute__((ext_vector_type(8))) float v8f;

__device__ __forceinline__ float sigm(float v) { return 1.0f / (1.0f + __expf(-v)); }

// ---- cooperative encode for one batch row (256 threads), results in LDS ----
__device__ __forceinline__ void encode_wg(
    int t,
    const float* __restrict__ xrow_g,
    const float* __restrict__ W1, const float* __restrict__ b1,
    const float* __restrict__ b2v, const float* __restrict__ b3,
    const float* __restrict__ W2g,
    float* sw2, float* xrow,
    float* c1v, float* c2v, float* c3v,
    float* s1, float* s2, float* s3)
{
    for (int i = t; i < C1N * C0N; i += 256) sw2[i] = W2g[i];
    for (int i = t; i < DIMN; i += 256)      xrow[i] = xrow_g[i];
    __syncthreads();
    if (t < C0N) {
        float acc = b1[t];
        const float* w1r = W1 + (size_t)t * DIMN;
        for (int i = 0; i < DIMN; ++i) acc += xrow[i] * w1r[i];
        float c = sigm(acc);
        c1v[t] = c; s1[t] = c * (1.0f - c);
    }
    __syncthreads();
    if (t < C1N) {
        float acc = b2v[t];
        for (int j = 0; j < C0N; ++j) acc += c1v[j] * sw2[t * C0N + j];
        float c = sigm(acc);
        c2v[t] = c; s2[t] = c * (1.0f - c);
    }
    __syncthreads();
    if (t < C0N) {
        float acc = b3[t];
        for (int l = 0; l < C1N; ++l) acc += c2v[l] * sw2[l * C0N + t];
        float c = sigm(acc);
        c3v[t] = c; s3[t] = c * (1.0f - c);
    }
    __syncthreads();
}

struct JacShared {
    float w2[C1N * C0N];                         // 8 KB
    float c1v[C0N], c2v[C1N], c3v[C0N];
    float s1[C0N],  s2[C1N],  s3[C0N];
    union { float xrow[DIMN]; float a3[C1N * DTILE]; } u;   // 16 KB
    float b2[C0N * B2STR];                       // 36 KB
};

// ---- main Jacobian kernel: one WG per (variant, b, d-tile) ----
__global__ __launch_bounds__(256) void jac_kernel(
    const float* __restrict__ x, const float* __restrict__ xn, const float* __restrict__ z,
    const float* __restrict__ W1, const float* __restrict__ b1,
    const float* __restrict__ W2, const float* __restrict__ b2v,
    const float* __restrict__ b3, float* __restrict__ out)
{
    __shared__ JacShared sm;
    const int t   = threadIdx.x;
    const int bid = blockIdx.x;
    const int dt  = bid & 7;                 // d-tile index (8 tiles of 128)
    const int b   = (bid >> 3) & (BATCHN - 1);
    const int v   = bid >> 10;               // 0:x 1:x_noise 2:z
    const float* src = (v == 0) ? x : (v == 1) ? xn : z;

    encode_wg(t, src + (size_t)b * DIMN, W1, b1, b2v, b3, W2,
              sm.w2, sm.u.xrow, sm.c1v, sm.c2v, sm.c3v, sm.s1, sm.s2, sm.s3);

    const int d0 = dt * DTILE;

    // ---- A3[j, dd] = sum_k W2[j,k] * s3[k] * W1[k, d0+dd]   (32 x 128) ----
    {
        const int dd = t & (DTILE - 1);
        const int jb = (t >> 7) * 16;        // lower/upper 16 rows
        float acc[16];
        #pragma unroll
        for (int q = 0; q < 16; ++q) acc[q] = 0.0f;
        for (int k = 0; k < C0N; ++k) {
            float w1v = W1[(size_t)k * DIMN + d0 + dd] * sm.s3[k];
            #pragma unroll
            for (int q = 0; q < 16; ++q)
                acc[q] += sm.w2[(jb + q) * C0N + k] * w1v;
        }
        #pragma unroll
        for (int q = 0; q < 16; ++q)
            sm.u.a3[(jb + q) * DTILE + dd] = acc[q];
    }
    __syncthreads();

    // ---- B2[j, dd] = s1[j] * sum_c W2[c,j] * s2[c] * A3[c, dd]   (64 x 128) ----
    {
        const int dd = t & (DTILE - 1);
        const int jb = (t >> 7) * 32;        // lower/upper 32 rows
        float acc[32];
        #pragma unroll
        for (int q = 0; q < 32; ++q) acc[q] = 0.0f;
        for (int c = 0; c < C1N; ++c) {
            float a3v = sm.u.a3[c * DTILE + dd] * sm.s2[c];
            #pragma unroll
            for (int q = 0; q < 32; ++q)
                acc[q] += sm.w2[c * C0N + jb + q] * a3v;
        }
        #pragma unroll
        for (int q = 0; q < 32; ++q)
            sm.b2[(jb + q) * B2STR + dd] = acc[q] * sm.s1[jb + q];
    }
    __syncthreads();

    // ---- J[b, i, d] = sum_j W1[j,i] * B2[j, d] via V_WMMA_F32_16X16X4_F32 ----
    const int w    = t >> 5;
    const int lane = t & 31;
    const int m    = lane & 15;              // M (A) / N (B,C)
    const int kh   = lane >> 4;              // K-half select
    float* __restrict__ jac = out + JACOFF + (size_t)v * JACELTS;

    for (int itl = 0; itl < 8; ++itl) {
        const int i0 = (w * 8 + itl) * 16;
        v2f afrag[16];
        #pragma unroll
        for (int kk = 0; kk < 16; ++kk) {    // A: 16x64 block, reused over d-subtiles
            const int j0 = 4 * kk + 2 * kh;
            afrag[kk].x = W1[(size_t)j0 * DIMN + i0 + m];
            afrag[kk].y = W1[(size_t)(j0 + 1) * DIMN + i0 + m];
        }
        for (int ds = 0; ds < 8; ++ds) {
            const int dd0 = ds * 16;
            v8f acc = {};
            #pragma unroll
            for (int kk = 0; kk < 16; ++kk) {
                const int j0 = 4 * kk + 2 * kh;
                v2f bfrag;
                bfrag.x = sm.b2[j0 * B2STR + dd0 + m];
                bfrag.y = sm.b2[(j0 + 1) * B2STR + dd0 + m];
                acc = __builtin_amdgcn_wmma_f32_16x16x4_f32(
                    false, afrag[kk], false, bfrag, (short)0, acc, false, false);
            }
            // scrambled layout: flat = (i*BATCH + b)*DIM + d ; non-temporal (1.6 GB stream)
            const size_t dcol = (size_t)(d0 + dd0 + m);
            #pragma unroll
            for (int r = 0; r < 8; ++r) {
                const int i = i0 + r + 8 * kh;
                float* p = jac + (size_t)(i * BATCHN + b) * DIMN + dcol;
                __builtin_nontemporal_store(acc[r], p);
            }
        }
    }
}

// ---- recover + c2 outputs (x variant only) ----
struct RecShared {
    float w2[C1N * C0N];
    float xrow[DIMN];
    float c1v[C0N], c2v[C1N], c3v[C0N];
    float s1[C0N],  s2[C1N],  s3[C0N];
};

__global__ __launch_bounds__(256) void recover_kernel(
    const float* __restrict__ x,
    const float* __restrict__ W1, const float* __restrict__ b1,
    const float* __restrict__ W2, const float* __restrict__ b2v,
    const float* __restrict__ b3, const float* __restrict__ b_r,
    float* __restrict__ out)
{
    __shared__ RecShared sm;
    const int t = threadIdx.x;
    const int b = blockIdx.x;

    encode_wg(t, x + (size_t)b * DIMN, W1, b1, b2v, b3, W2,
              sm.w2, sm.xrow, sm.c1v, sm.c2v, sm.c3v, sm.s1, sm.s2, sm.s3);

    #pragma unroll
    for (int q = 0; q < 4; ++q) {
        const int i = t + q * 256;
        float acc = b_r[i];
        for (int k = 0; k < C0N; ++k)
            acc += sm.c3v[k] * W1[(size_t)k * DIMN + i];
        out[(size_t)b * DIMN + i] = sigm(acc);
    }
    if (t < C1N)
        out[(size_t)BATCHN * DIMN + b * C1N + t] = sm.c2v[t];
}

extern "C" void kernel_launch(void* const* d_in, const int* in_sizes, int n_in,
                              void* d_out, int out_size, void* d_ws, size_t ws_size,
                              hipStream_t stream) {
    (void)in_sizes; (void)n_in; (void)d_ws; (void)ws_size; (void)out_size;
    const float* x   = (const float*)d_in[0];
    const float* xn  = (const float*)d_in[1];
    const float* z   = (const float*)d_in[2];
    const float* W1  = (const float*)d_in[3];
    const float* b1  = (const float*)d_in[4];
    const float* W2  = (const float*)d_in[5];
    const float* b2  = (const float*)d_in[6];
    const float* b3  = (const float*)d_in[7];
    const float* b_r = (const float*)d_in[8];
    float* out = (float*)d_out;

    recover_kernel<<<BATCHN, 256, 0, stream>>>(x, W1, b1, W2, b2, b3, b_r, out);
    jac_kernel<<<3 * BATCHN * (DIMN / DTILE), 256, 0, stream>>>(
        x, xn, z, W1, b1, W2, b2, b3, out);
}